// BenchGNN_29300266893894
// MI455X (gfx1250) — compile-verified
//
#include <hip/hip_runtime.h>
#include <hip/hip_bf16.h>
#include <math.h>

typedef float v2f __attribute__((ext_vector_type(2)));
typedef float v8f __attribute__((ext_vector_type(8)));

#define FIN 128
#define HID 96
#define GNUM 512

// ---------------- utility kernels ----------------

__global__ void fill_f32(float* __restrict__ p, float v, long n) {
  long i = (long)blockIdx.x * blockDim.x + threadIdx.x;
  if (i < n) p[i] = v;
}

__global__ void count_nodes(const int* __restrict__ batch, float* __restrict__ cnt, int N) {
  int i = blockIdx.x * blockDim.x + threadIdx.x;
  if (i < N) atomicAdd(&cnt[batch[i]], 1.0f);
}

__global__ void pool_sum(const float* __restrict__ x, const int* __restrict__ batch,
                         float* __restrict__ gsum, int N, int F) {
  long i = (long)blockIdx.x * blockDim.x + threadIdx.x;
  if (i >= (long)N * F) return;
  int n = (int)(i / F), f = (int)(i % F);
  atomicAdd(&gsum[(long)batch[n] * F + f], x[i]);
}

__device__ inline void atomicMaxF(float* addr, float v) {
  // works for mixed signs: int-order == float-order for v>=0,
  // unsigned-order is reversed float-order for v<0.
  if (v >= 0.0f) atomicMax((int*)addr, __float_as_int(v));
  else           atomicMin((unsigned int*)addr, __float_as_uint(v));
}

__global__ void pool_sum_max(const float* __restrict__ x, const int* __restrict__ batch,
                             float* __restrict__ gsum, float* __restrict__ gmax, int N, int F) {
  long i = (long)blockIdx.x * blockDim.x + threadIdx.x;
  if (i >= (long)N * F) return;
  int n = (int)(i / F), f = (int)(i % F);
  long gi = (long)batch[n] * F + f;
  float v = x[i];
  atomicAdd(&gsum[gi], v);
  atomicMaxF(&gmax[gi], v);
}

__global__ void gap_div(float* __restrict__ gsum, const float* __restrict__ cnt, int G, int F) {
  int i = blockIdx.x * blockDim.x + threadIdx.x;
  if (i >= G * F) return;
  gsum[i] /= fmaxf(cnt[i / F], 1.0f);
}

__global__ void deg_kernel(const int* __restrict__ dst, float* __restrict__ deg, int E) {
  int e = blockIdx.x * blockDim.x + threadIdx.x;
  if (e < E) atomicAdd(&deg[dst[e]], 1.0f);
}

__global__ void dinv_kernel(const float* __restrict__ deg, float* __restrict__ dinv, int N) {
  int i = blockIdx.x * blockDim.x + threadIdx.x;
  if (i < N) dinv[i] = rsqrtf(deg[i] + 1.0f);
}

// one wave (32 lanes) per edge; 96 features -> 3 per lane
__global__ void scatter_edges(const float* __restrict__ h, const int* __restrict__ src,
                              const int* __restrict__ dst, const float* __restrict__ dinv,
                              float* __restrict__ agg, int E, int F) {
  long t = (long)blockIdx.x * blockDim.x + threadIdx.x;
  int e = (int)(t >> 5);
  int lane = (int)(t & 31);
  if (e >= E) return;
  int s = src[e], d = dst[e];
  float w = dinv[s] * dinv[d];
  const float* hs = h + (long)s * F;
  float* ad = agg + (long)d * F;
  for (int f = lane; f < F; f += 32)
    atomicAdd(&ad[f], hs[f] * w);
}

__global__ void finalize_kernel(const float* __restrict__ agg, const float* __restrict__ h,
                                const float* __restrict__ dinv, const float* __restrict__ bias,
                                float* __restrict__ xout, int N, int F, int relu) {
  long i = (long)blockIdx.x * blockDim.x + threadIdx.x;
  if (i >= (long)N * F) return;
  int n = (int)(i / F), f = (int)(i % F);
  float di = dinv[n];
  float v = agg[i] + h[i] * di * di + bias[f];
  if (relu) v = fmaxf(v, 0.0f);
  xout[i] = v;
}

__global__ void accum_readout(const float* __restrict__ gmax, const float* __restrict__ gap,
                              float* __restrict__ hacc, int G, int F) {
  int i = blockIdx.x * blockDim.x + threadIdx.x;
  if (i >= G * F) return;
  int g = i / F, f = i % F;
  float m = gmax[i];
  if (m == -INFINITY) m = 0.0f;  // jnp.where(isneginf, 0, m)
  hacc[(long)g * 2 * F + f]     += m;
  hacc[(long)g * 2 * F + F + f] += gap[i];
}

__global__ void final_linear(const float* __restrict__ A, const float* __restrict__ W,
                             const float* __restrict__ b, float* __restrict__ out,
                             int M, int K, int NO) {
  int i = blockIdx.x * blockDim.x + threadIdx.x;
  if (i >= M * NO) return;
  int m = i / NO, o = i % NO;
  float s = b[o];
  const float* a = A + (long)m * K;
  for (int k = 0; k < K; ++k) s += a[k] * W[k * NO + o];
  out[i] = s;
}

// ---------------- WMMA f32 GEMM ----------------
// C[M x NC] = A' @ B (+bias, optional relu)
// A' row n = CONCAT ? [A[n, 0:Ka] | Ag[batch[n], 0:Kg]] : A[n, 0:Ka]
// One wave computes a 16-row x (NT*16)-col strip via V_WMMA_F32_16X16X4_F32.
// Requires M % 16 == 0 (true here: 50000, 512), NC == NT*16, K % 4 == 0.
template <int NT, bool CONCAT>
__global__ __launch_bounds__(256) void gemm_wmma_kernel(
    const float* __restrict__ A, const float* __restrict__ Ag, const int* __restrict__ batch,
    const float* __restrict__ B, const float* __restrict__ bias, float* __restrict__ C,
    int M, int Ka, int Kg, int NC, int relu) {
  const int lane = threadIdx.x & 31;
  const int wave = threadIdx.x >> 5;
  const int rowtile = blockIdx.x * (blockDim.x >> 5) + wave;
  if (rowtile * 16 >= M) return;  // whole-wave exit: EXEC stays all-ones for WMMA

  const int halfsel = lane >> 4;   // lanes 0-15 -> K pair {0,1}; lanes 16-31 -> {2,3}
  const int l16 = lane & 15;
  const int row = rowtile * 16 + l16;  // A row owned by this lane
  const int K = Ka + (CONCAT ? Kg : 0);

  const float* arow = A + (long)row * Ka;
  const float* grow = CONCAT ? (Ag + (long)batch[row] * Kg) : nullptr;

  v8f acc[NT];
#pragma unroll
  for (int t = 0; t < NT; ++t)
#pragma unroll
    for (int r = 0; r < 8; ++r) acc[t][r] = 0.0f;

  for (int k = 0; k < K; k += 4) {
    const int kk = k + halfsel * 2;
    v2f a;
    if (CONCAT) {
      a.x = (kk     < Ka) ? arow[kk]     : grow[kk - Ka];
      a.y = (kk + 1 < Ka) ? arow[kk + 1] : grow[kk + 1 - Ka];
    } else {
      a.x = arow[kk];
      a.y = arow[kk + 1];
    }
    const float* b0 = B + (long)kk * NC + l16;  // B[kk][col], B[kk+1][col]
#pragma unroll
    for (int t = 0; t < NT; ++t) {
      v2f b;
      b.x = b0[t * 16];
      b.y = b0[t * 16 + NC];
      acc[t] = __builtin_amdgcn_wmma_f32_16x16x4_f32(
          false, a, false, b, (short)0, acc[t], false, false);
    }
  }

  // D layout: VGPR r -> row r (lanes 0-15) / row r+8 (lanes 16-31); col = lane&15
  const int rbase = rowtile * 16 + halfsel * 8;
#pragma unroll
  for (int t = 0; t < NT; ++t) {
    const int col = t * 16 + l16;
    const float bv = bias ? bias[col] : 0.0f;
#pragma unroll
    for (int r = 0; r < 8; ++r) {
      float v = acc[t][r] + bv;
      if (relu) v = fmaxf(v, 0.0f);
      C[(long)(rbase + r) * NC + col] = v;
    }
  }
}

// ---------------- host orchestration ----------------

static inline int cdiv_l(long a, long b) { return (int)((a + b - 1) / b); }

extern "C" void kernel_launch(void* const* d_in, const int* in_sizes, int n_in,
                              void* d_out, int out_size, void* d_ws, size_t ws_size,
                              hipStream_t stream) {
  (void)n_in; (void)out_size; (void)ws_size;
  const float* x0  = (const float*)d_in[0];
  const int*  eidx = (const int*)d_in[1];
  const int*  batch= (const int*)d_in[2];
  const float* W1  = (const float*)d_in[3];
  const float* b1  = (const float*)d_in[4];
  const float* W2  = (const float*)d_in[5];
  const float* b2  = (const float*)d_in[6];
  const float* W3  = (const float*)d_in[7];
  const float* b3  = (const float*)d_in[8];
  const float* Wl1 = (const float*)d_in[9];
  const float* bl1 = (const float*)d_in[10];
  const float* Wl2 = (const float*)d_in[11];
  const float* bl2 = (const float*)d_in[12];
  const float* Wl3 = (const float*)d_in[13];
  const float* bl3 = (const float*)d_in[14];

  const int N = in_sizes[0] / FIN;   // 50000
  const int E = in_sizes[1] / 2;     // 800000
  const int G = GNUM;                // 512
  const int* src = eidx;
  const int* dst = eidx + E;

  // workspace layout (floats)
  float* p = (float*)d_ws;
  float* gap0 = p; p += (long)G * FIN;       // mean pool of x0 (sum -> in-place div)
  float* cnt  = p; p += G;
  float* deg  = p; p += N;
  float* dinv = p; p += N;
  float* h    = p; p += (long)N * HID;       // GEMM output (pre-aggregation)
  float* agg  = p; p += (long)N * HID;       // scatter accumulator
  float* xcur = p; p += (long)N * HID;       // node features after each layer
  float* gsum = p; p += (long)G * HID;       // per-layer mean pool (sum -> gap)
  float* gmax = p; p += (long)G * HID;       // per-layer max pool
  float* hacc = p; p += (long)G * 2 * HID;   // x1 + x2 + x3
  float* m1   = p; p += (long)G * HID;
  float* m2   = p; p += (long)G * (HID / 2);

  const int T = 256;
  const int WPB = 8;  // waves per block in GEMM

  // ---- init + degree + initial mean pool ----
  fill_f32<<<cdiv_l((long)G * FIN, T), T, 0, stream>>>(gap0, 0.0f, (long)G * FIN);
  fill_f32<<<cdiv_l(G, T), T, 0, stream>>>(cnt, 0.0f, G);
  fill_f32<<<cdiv_l(N, T), T, 0, stream>>>(deg, 0.0f, N);
  fill_f32<<<cdiv_l((long)G * 2 * HID, T), T, 0, stream>>>(hacc, 0.0f, (long)G * 2 * HID);

  count_nodes<<<cdiv_l(N, T), T, 0, stream>>>(batch, cnt, N);
  pool_sum<<<cdiv_l((long)N * FIN, T), T, 0, stream>>>(x0, batch, gap0, N, FIN);
  gap_div<<<cdiv_l((long)G * FIN, T), T, 0, stream>>>(gap0, cnt, G, FIN);
  deg_kernel<<<cdiv_l(E, T), T, 0, stream>>>(dst, deg, E);
  dinv_kernel<<<cdiv_l(N, T), T, 0, stream>>>(deg, dinv, N);

  const int rowtilesN = N / 16;          // 3125
  const int gemmBlkN = cdiv_l(rowtilesN, WPB);
  const int rowtilesG = G / 16;          // 32
  const int gemmBlkG = cdiv_l(rowtilesG, WPB);

  // ---- layer 1: concat(x0, gap0[batch]) @ W1 ; aggregate ; relu ; pool ----
  gemm_wmma_kernel<6, true><<<gemmBlkN, 256, 0, stream>>>(
      x0, gap0, batch, W1, nullptr, h, N, FIN, FIN, HID, 0);
  fill_f32<<<cdiv_l((long)N * HID, T), T, 0, stream>>>(agg, 0.0f, (long)N * HID);
  scatter_edges<<<cdiv_l((long)E * 32, T), T, 0, stream>>>(h, src, dst, dinv, agg, E, HID);
  finalize_kernel<<<cdiv_l((long)N * HID, T), T, 0, stream>>>(agg, h, dinv, b1, xcur, N, HID, 1);
  fill_f32<<<cdiv_l((long)G * HID, T), T, 0, stream>>>(gsum, 0.0f, (long)G * HID);
  fill_f32<<<cdiv_l((long)G * HID, T), T, 0, stream>>>(gmax, -INFINITY, (long)G * HID);
  pool_sum_max<<<cdiv_l((long)N * HID, T), T, 0, stream>>>(xcur, batch, gsum, gmax, N, HID);
  gap_div<<<cdiv_l((long)G * HID, T), T, 0, stream>>>(gsum, cnt, G, HID);
  accum_readout<<<cdiv_l((long)G * HID, T), T, 0, stream>>>(gmax, gsum, hacc, G, HID);

  // ---- layer 2 (gsum currently holds gap(x_layer1) == x_glob for layer 2) ----
  gemm_wmma_kernel<6, true><<<gemmBlkN, 256, 0, stream>>>(
      xcur, gsum, batch, W2, nullptr, h, N, HID, HID, HID, 0);
  fill_f32<<<cdiv_l((long)N * HID, T), T, 0, stream>>>(agg, 0.0f, (long)N * HID);
  scatter_edges<<<cdiv_l((long)E * 32, T), T, 0, stream>>>(h, src, dst, dinv, agg, E, HID);
  finalize_kernel<<<cdiv_l((long)N * HID, T), T, 0, stream>>>(agg, h, dinv, b2, xcur, N, HID, 1);
  fill_f32<<<cdiv_l((long)G * HID, T), T, 0, stream>>>(gsum, 0.0f, (long)G * HID);
  fill_f32<<<cdiv_l((long)G * HID, T), T, 0, stream>>>(gmax, -INFINITY, (long)G * HID);
  pool_sum_max<<<cdiv_l((long)N * HID, T), T, 0, stream>>>(xcur, batch, gsum, gmax, N, HID);
  gap_div<<<cdiv_l((long)G * HID, T), T, 0, stream>>>(gsum, cnt, G, HID);
  accum_readout<<<cdiv_l((long)G * HID, T), T, 0, stream>>>(gmax, gsum, hacc, G, HID);

  // ---- layer 3 (no relu on conv output) ----
  gemm_wmma_kernel<6, true><<<gemmBlkN, 256, 0, stream>>>(
      xcur, gsum, batch, W3, nullptr, h, N, HID, HID, HID, 0);
  fill_f32<<<cdiv_l((long)N * HID, T), T, 0, stream>>>(agg, 0.0f, (long)N * HID);
  scatter_edges<<<cdiv_l((long)E * 32, T), T, 0, stream>>>(h, src, dst, dinv, agg, E, HID);
  finalize_kernel<<<cdiv_l((long)N * HID, T), T, 0, stream>>>(agg, h, dinv, b3, xcur, N, HID, 0);
  fill_f32<<<cdiv_l((long)G * HID, T), T, 0, stream>>>(gsum, 0.0f, (long)G * HID);
  fill_f32<<<cdiv_l((long)G * HID, T), T, 0, stream>>>(gmax, -INFINITY, (long)G * HID);
  pool_sum_max<<<cdiv_l((long)N * HID, T), T, 0, stream>>>(xcur, batch, gsum, gmax, N, HID);
  gap_div<<<cdiv_l((long)G * HID, T), T, 0, stream>>>(gsum, cnt, G, HID);
  accum_readout<<<cdiv_l((long)G * HID, T), T, 0, stream>>>(gmax, gsum, hacc, G, HID);

  // ---- readout MLP: [512,192]@[192,96] relu ; @[96,48] relu ; @[48,10] ----
  gemm_wmma_kernel<6, false><<<gemmBlkG, 256, 0, stream>>>(
      hacc, nullptr, nullptr, Wl1, bl1, m1, G, 2 * HID, 0, HID, 1);
  gemm_wmma_kernel<3, false><<<gemmBlkG, 256, 0, stream>>>(
      m1, nullptr, nullptr, Wl2, bl2, m2, G, HID, 0, HID / 2, 1);
  final_linear<<<cdiv_l((long)G * 10, T), T, 0, stream>>>(
      m2, Wl3, bl3, (float*)d_out, G, HID / 2, 10);
}